// SemalinkAttentiveAggregator_38646115729709
// MI455X (gfx1250) — compile-verified
//
#include <hip/hip_runtime.h>
#include <hip/hip_bf16.h>
#include <math.h>

// ---------------------------------------------------------------------------
// SemalinkAttentiveAggregator for MI455X (gfx1250, wave32)
// fp32 WMMA (v_wmma_f32_16x16x4_f32) for all GEMMs; L2-resident edge passes.
// ---------------------------------------------------------------------------

typedef __attribute__((ext_vector_type(2))) float v2f;
typedef __attribute__((ext_vector_type(8))) float v8f;

#define N_NODES      50000
#define N_HYPER      20000
#define N_EDGES      1000000
#define N_REL        256
#define DIM          128
#define SEM_DIM      64
#define LRELU        0.01f
#define LN_EPS_F     1e-5f

__device__ __forceinline__ v8f wmma_f32x4(v2f a, v2f b, v8f c) {
  // D = A(16x4,f32) * B(4x16,f32) + C(16x16,f32)
  return __builtin_amdgcn_wmma_f32_16x16x4_f32(
      /*neg_a=*/false, a, /*neg_b=*/false, b,
      /*c_mod=*/(short)0, c, /*reuse_a=*/false, /*reuse_b=*/false);
}

__device__ __forceinline__ void atomicMaxFloat(float* addr, float v) {
  if (v >= 0.0f) atomicMax((int*)addr, __float_as_int(v));
  else           atomicMin((unsigned int*)addr, __float_as_uint(v));
}

// ---------------------------------------------------------------------------
// init: seg_max = -inf, denom = 0, agg = 0
// ---------------------------------------------------------------------------
__global__ void k_init(float* __restrict__ seg_max, float* __restrict__ denom,
                       float* __restrict__ agg) {
  int i = blockIdx.x * blockDim.x + threadIdx.x;
  if (i < N_HYPER * DIM) agg[i] = 0.0f;
  if (i < N_HYPER) { seg_max[i] = -INFINITY; denom[i] = 0.0f; }
}

// ---------------------------------------------------------------------------
// sem_t[256 x 128] = sem_emb[256 x 64] @ sem_w[128 x 64]^T + sem_b
// One wave -> one 16x16 tile. grid = 256/16 blocks, 8 waves cover N=128.
// ---------------------------------------------------------------------------
__global__ void k_gemm_sem(const float* __restrict__ E, const float* __restrict__ W,
                           const float* __restrict__ bias, float* __restrict__ C) {
  const int wave = threadIdx.x >> 5;
  const int lane = threadIdx.x & 31;
  const int hi = lane >> 4, lo = lane & 15;
  const int rbase = blockIdx.x * 16;
  const int nb = wave * 16;

  v8f acc = {};
  const float* arow = E + (size_t)(rbase + lo) * SEM_DIM + 2 * hi;   // A[m][k]
  const float* brow = W + (size_t)(nb + lo) * SEM_DIM + 2 * hi;      // B[k][n]=W[n][k]
#pragma unroll
  for (int k0 = 0; k0 < SEM_DIM; k0 += 4) {
    v2f a = *(const v2f*)(arow + k0);
    v2f b = *(const v2f*)(brow + k0);
    acc = wmma_f32x4(a, b, acc);
  }
  const float bn = bias[nb + lo];
#pragma unroll
  for (int i = 0; i < 8; ++i) {
    int r = rbase + i + 8 * hi;
    C[(size_t)r * DIM + nb + lo] = acc[i] + bn;
  }
}

// ---------------------------------------------------------------------------
// C[M x 128] = A[M x 128] @ B[128 x 128]   (B row-major, B[k][n] = W_r[k][n])
// grid = M/16 blocks (M divisible by 16 here), 8 waves cover N=128.
// ---------------------------------------------------------------------------
__global__ void k_gemm_wr(const float* __restrict__ A, const float* __restrict__ B,
                          float* __restrict__ C, int M) {
  const int wave = threadIdx.x >> 5;
  const int lane = threadIdx.x & 31;
  const int hi = lane >> 4, lo = lane & 15;
  const int rbase = blockIdx.x * 16;
  const int nb = wave * 16;
  if (rbase >= M) return;                   // wave-uniform

  int row = rbase + lo;
  if (row >= M) row = M - 1;                // clamp (loads only)

  v8f acc = {};
  const float* arow = A + (size_t)row * DIM + 2 * hi;
#pragma unroll
  for (int k0 = 0; k0 < DIM; k0 += 4) {
    v2f a = *(const v2f*)(arow + k0);
    v2f b;
    b.x = B[(size_t)(k0 + 2 * hi) * DIM + nb + lo];
    b.y = B[(size_t)(k0 + 2 * hi + 1) * DIM + nb + lo];
    acc = wmma_f32x4(a, b, acc);
  }
#pragma unroll
  for (int i = 0; i < 8; ++i) {
    int r = rbase + i + 8 * hi;
    if (r < M) C[(size_t)r * DIM + nb + lo] = acc[i];
  }
}

// ---------------------------------------------------------------------------
// Edge pass 1: score[e] = dot(r_h[node], tanh(r_t[edge] + sem_t[rel]))
//              seg_max[edge] = max(...)
// One wave per edge; lane handles 4 dims (float4). 8 edges per 256-thr block.
// ---------------------------------------------------------------------------
__global__ void k_edge_score(const int* __restrict__ sl,
                             const float* __restrict__ r_h,
                             const float* __restrict__ r_t,
                             const float* __restrict__ sem_t,
                             float* __restrict__ scores,
                             float* __restrict__ seg_max) {
  const int e = blockIdx.x * 8 + (threadIdx.x >> 5);
  if (e >= N_EDGES) return;                 // wave-uniform
  const int lane = threadIdx.x & 31;

  const int ni = sl[3 * e + 0];
  const int ei = sl[3 * e + 1];
  const int ri = sl[3 * e + 2];

  const float4 a  = ((const float4*)(r_h   + (size_t)ni * DIM))[lane];
  const float4 t4 = ((const float4*)(r_t   + (size_t)ei * DIM))[lane];
  const float4 s4 = ((const float4*)(sem_t + (size_t)ri * DIM))[lane];

  float p = a.x * tanhf(t4.x + s4.x) + a.y * tanhf(t4.y + s4.y)
          + a.z * tanhf(t4.z + s4.z) + a.w * tanhf(t4.w + s4.w);
  p += __shfl_xor(p, 16, 32);
  p += __shfl_xor(p,  8, 32);
  p += __shfl_xor(p,  4, 32);
  p += __shfl_xor(p,  2, 32);
  p += __shfl_xor(p,  1, 32);

  if (lane == 0) {
    scores[e] = p;
    atomicMaxFloat(&seg_max[ei], p);
  }
}

// ---------------------------------------------------------------------------
// Edge pass 2: ex = exp(score - seg_max[edge]); denom[edge] += ex;
//              agg[edge] += ex * node_emb[node]    (normalized later)
// ---------------------------------------------------------------------------
__global__ void k_edge_agg(const int* __restrict__ sl,
                           const float* __restrict__ scores,
                           const float* __restrict__ seg_max,
                           const float* __restrict__ node_emb,
                           float* __restrict__ denom,
                           float* __restrict__ agg) {
  const int e = blockIdx.x * 8 + (threadIdx.x >> 5);
  if (e >= N_EDGES) return;                 // wave-uniform
  const int lane = threadIdx.x & 31;

  const int ni = sl[3 * e + 0];
  const int ei = sl[3 * e + 1];

  const float ex = __expf(scores[e] - seg_max[ei]);
  if (lane == 0) atomicAdd(&denom[ei], ex);

  const float4 x = ((const float4*)(node_emb + (size_t)ni * DIM))[lane];
  float* ap = agg + (size_t)ei * DIM + lane * 4;
  atomicAdd(ap + 0, ex * x.x);
  atomicAdd(ap + 1, ex * x.y);
  atomicAdd(ap + 2, ex * x.z);
  atomicAdd(ap + 3, ex * x.w);
}

// ---------------------------------------------------------------------------
// Output: out = LayerNorm(LeakyReLU((agg/denom) @ lin_w^T + lin_b))
// Each wave owns 16 FULL rows (8 column tiles, 64 acc VGPRs) so LayerNorm
// reduces within 16-lane halves via shfl_xor. 128 rows per 256-thr block.
// ---------------------------------------------------------------------------
__global__ void k_out(const float* __restrict__ agg, const float* __restrict__ denom,
                      const float* __restrict__ lw, const float* __restrict__ lb,
                      const float* __restrict__ gamma, const float* __restrict__ beta,
                      float* __restrict__ out) {
  const int wave = threadIdx.x >> 5;
  const int lane = threadIdx.x & 31;
  const int hi = lane >> 4, lo = lane & 15;
  const int rbase = blockIdx.x * 128 + wave * 16;
  if (rbase >= N_HYPER) return;             // wave-uniform

  int row = rbase + lo;
  if (row >= N_HYPER) row = N_HYPER - 1;    // clamp (loads only)
  const float inv_d = 1.0f / denom[row];

  v8f zero = {};
  v8f acc[8];
#pragma unroll
  for (int t = 0; t < 8; ++t) acc[t] = zero;

  const float* arow = agg + (size_t)row * DIM + 2 * hi;
#pragma unroll
  for (int k0 = 0; k0 < DIM; k0 += 4) {
    v2f a = *(const v2f*)(arow + k0);
    a.x *= inv_d; a.y *= inv_d;
#pragma unroll
    for (int t = 0; t < 8; ++t) {
      // B[k][n] = lin_w[n][k] -> contiguous in k -> 8-byte load
      v2f b = *(const v2f*)(lw + (size_t)(t * 16 + lo) * DIM + k0 + 2 * hi);
      acc[t] = wmma_f32x4(a, b, acc[t]);
    }
  }

  // Epilogue: VGPR i holds row (rbase + i + 8*hi), column (t*16 + lo).
#pragma unroll
  for (int i = 0; i < 8; ++i) {
    const int r = rbase + i + 8 * hi;
    float vals[8];
    float s = 0.0f, ss = 0.0f;
#pragma unroll
    for (int t = 0; t < 8; ++t) {
      float v = acc[t][i] + lb[t * 16 + lo];
      v = (v >= 0.0f) ? v : LRELU * v;
      vals[t] = v;
      s += v; ss += v * v;
    }
    // reduce across the 16-lane half that owns this row
    s  += __shfl_xor(s, 1, 32);  s  += __shfl_xor(s, 2, 32);
    s  += __shfl_xor(s, 4, 32);  s  += __shfl_xor(s, 8, 32);
    ss += __shfl_xor(ss, 1, 32); ss += __shfl_xor(ss, 2, 32);
    ss += __shfl_xor(ss, 4, 32); ss += __shfl_xor(ss, 8, 32);

    const float mean = s * (1.0f / 128.0f);
    const float var  = ss * (1.0f / 128.0f) - mean * mean;
    const float rs   = rsqrtf(var + LN_EPS_F);

    if (r < N_HYPER) {
#pragma unroll
      for (int t = 0; t < 8; ++t) {
        const int n = t * 16 + lo;
        out[(size_t)r * DIM + n] = (vals[t] - mean) * rs * gamma[n] + beta[n];
      }
    }
  }
}

// ---------------------------------------------------------------------------
// launch
// ---------------------------------------------------------------------------
extern "C" void kernel_launch(void* const* d_in, const int* in_sizes, int n_in,
                              void* d_out, int out_size, void* d_ws, size_t ws_size,
                              hipStream_t stream) {
  const float* node_emb = (const float*)d_in[0];   // 50000 x 128
  const float* sem_emb  = (const float*)d_in[1];   // 256 x 64
  const float* hyp_emb  = (const float*)d_in[2];   // 20000 x 128
  const float* W_r      = (const float*)d_in[3];   // 128 x 128
  const float* lin_w    = (const float*)d_in[4];   // 128 x 128
  const float* lin_b    = (const float*)d_in[5];   // 128
  const float* sem_w    = (const float*)d_in[6];   // 128 x 64
  const float* sem_b    = (const float*)d_in[7];   // 128
  const float* ln_g     = (const float*)d_in[8];   // 128
  const float* ln_b     = (const float*)d_in[9];   // 128
  const int*   sl       = (const int*)d_in[10];    // 1M x 3
  float* out = (float*)d_out;

  // workspace layout (floats)
  float* ws = (float*)d_ws;
  float* sem_t   = ws;                                  // 256*128
  float* r_h     = sem_t + (size_t)N_REL * DIM;         // 50000*128
  float* r_t     = r_h + (size_t)N_NODES * DIM;         // 20000*128
  float* scores  = r_t + (size_t)N_HYPER * DIM;         // 1M
  float* seg_max = scores + (size_t)N_EDGES;            // 20000
  float* denom   = seg_max + (size_t)N_HYPER;           // 20000
  float* agg     = denom + (size_t)N_HYPER;             // 20000*128

  k_init<<<(N_HYPER * DIM + 255) / 256, 256, 0, stream>>>(seg_max, denom, agg);

  k_gemm_sem<<<N_REL / 16, 256, 0, stream>>>(sem_emb, sem_w, sem_b, sem_t);
  k_gemm_wr<<<N_NODES / 16, 256, 0, stream>>>(node_emb, W_r, r_h, N_NODES);
  k_gemm_wr<<<N_HYPER / 16, 256, 0, stream>>>(hyp_emb, W_r, r_t, N_HYPER);

  k_edge_score<<<(N_EDGES + 7) / 8, 256, 0, stream>>>(sl, r_h, r_t, sem_t,
                                                      scores, seg_max);
  k_edge_agg<<<(N_EDGES + 7) / 8, 256, 0, stream>>>(sl, scores, seg_max,
                                                    node_emb, denom, agg);

  k_out<<<(N_HYPER + 127) / 128, 256, 0, stream>>>(agg, denom, lin_w, lin_b,
                                                   ln_g, ln_b, out);
}